// FNO1d_25795573580567
// MI455X (gfx1250) — compile-verified
//
#include <hip/hip_runtime.h>
#include <math.h>

// ---------------- problem constants ----------------
constexpr int B_     = 64;
constexpr int T_     = 4096;
constexpr int W_     = 64;
constexpr int MODES_ = 32;
constexpr int NL_    = 4;
constexpr int OUT_T_ = 4096;
constexpr int R_     = B_ * W_;      // 4096 flattened (b, channel) rows

typedef float v2f __attribute__((ext_vector_type(2)));
typedef float v8f __attribute__((ext_vector_type(8)));

// CDNA5 native fp32 WMMA: D(16x16) = A(16x4) * B(4x16) + C
__device__ __forceinline__ v8f wmma4(v2f a, v2f b, v8f c) {
    return __builtin_amdgcn_wmma_f32_16x16x4_f32(
        /*neg_a=*/false, a, /*neg_b=*/false, b,
        /*c_mod=*/(short)0, c, /*reuse_a=*/false, /*reuse_b=*/false);
}

__device__ __forceinline__ float gelu_exact(float x) {
    // torch GELU default (erf form)
    return 0.5f * x * (1.0f + erff(x * 0.70710678118654752440f));
}

// ---------------- K0: Fourier basis ----------------
// F[t,m]   : m<32 -> cos(2pi m t/T), m>=32 -> -sin(2pi (m-32) t/T)        (T x 64)
// G[k,t]   : k<32 -> c_k cos(2pi k t/T), k>=32 -> -c_k sin(2pi (k-32)t/T) (64 x T)
//            c_0 = 1/T, c_k = 2/T  (row k=32 is all zeros since sin(0)=0)
__global__ void fno_basis_kernel(float* __restrict__ F, float* __restrict__ G) {
    int idx = blockIdx.x * blockDim.x + threadIdx.x;   // T_*64 threads
    int t = idx >> 6;
    int m = idx & 63;
    int mm = (m < MODES_) ? m : (m - MODES_);
    float ang = (2.0f * 3.14159265358979323846f * (float)mm * (float)t) / (float)T_;
    float s, c;
    sincosf(ang, &s, &c);
    float fv = (m < MODES_) ? c : -s;
    F[(size_t)t * 64 + m] = fv;
    float coef = (mm == 0) ? (1.0f / (float)T_) : (2.0f / (float)T_);
    float gv = (m < MODES_) ? (coef * c) : (-coef * s);
    G[(size_t)m * T_ + t] = gv;
}

// ---------------- K1: lift  h[b,w,t] = x[b,t]*lift_w[w] + lift_b[w] ----------------
__global__ void fno_lift_kernel(const float* __restrict__ x,
                                const float* __restrict__ lw,
                                const float* __restrict__ lb,
                                float* __restrict__ h) {
    size_t idx = (size_t)blockIdx.x * blockDim.x + threadIdx.x;  // B*W*T
    int t = (int)(idx & (T_ - 1));
    int w = (int)((idx >> 12) & (W_ - 1));
    int b = (int)(idx >> 18);
    h[idx] = x[(size_t)b * T_ + t] * lw[w] + lb[w];
}

// ---------------- K2: forward truncated DFT (WMMA) ----------------
// XF[r, 0:64] = h[r, :] (1x4096) @ F (4096x64).  One wave -> 16 rows x 64 cols.
__global__ void fno_dft_kernel(const float* __restrict__ h,
                               const float* __restrict__ F,
                               float* __restrict__ XF) {
    int wave = (blockIdx.x * blockDim.x + threadIdx.x) >> 5;
    int lane = threadIdx.x & 31;
    int half = lane >> 4;
    int lid  = lane & 15;
    int rowBase = wave * 16;                 // 256 waves cover R_=4096 rows

    v8f c0 = {}, c1 = {}, c2 = {}, c3 = {};
    const float* arow = h + (size_t)(rowBase + lid) * T_;
    for (int k = 0; k < T_; k += 4) {
        int kk = k + 2 * half;
        v2f a;
        a.x = arow[kk];
        a.y = arow[kk + 1];
        const float* f0 = F + (size_t)kk * 64;
        const float* f1 = f0 + 64;
        v2f b0, b1, b2, b3;
        b0.x = f0[lid];      b0.y = f1[lid];
        b1.x = f0[16 + lid]; b1.y = f1[16 + lid];
        b2.x = f0[32 + lid]; b2.y = f1[32 + lid];
        b3.x = f0[48 + lid]; b3.y = f1[48 + lid];
        c0 = wmma4(a, b0, c0);
        c1 = wmma4(a, b1, c1);
        c2 = wmma4(a, b2, c2);
        c3 = wmma4(a, b3, c3);
    }
    #pragma unroll
    for (int r = 0; r < 8; ++r) {
        size_t row = (size_t)(rowBase + r + 8 * half);
        XF[row * 64 +      lid] = c0[r];
        XF[row * 64 + 16 + lid] = c1[r];
        XF[row * 64 + 32 + lid] = c2[r];
        XF[row * 64 + 48 + lid] = c3[r];
    }
}

// ---------------- K3: complex spectral mixing ----------------
// Z[b,o,m] = sum_i XF[b,i,m] * (wr[i,o,m] + j wi[i,o,m]);  Z stored [ (b*W+o)*64 + m | +32 ]
__global__ void fno_specmul_kernel(const float* __restrict__ XF,
                                   const float* __restrict__ wr,   // [W,W,MODES] for layer l
                                   const float* __restrict__ wi,
                                   float* __restrict__ Z) {
    int idx = blockIdx.x * blockDim.x + threadIdx.x;   // B*W*MODES threads
    int m = idx & (MODES_ - 1);
    int o = (idx >> 5) & (W_ - 1);
    int b = idx >> 11;
    float zr = 0.0f, zi = 0.0f;
    const float* xfb = XF + (size_t)b * W_ * 64;
    for (int i = 0; i < W_; ++i) {
        float xr = xfb[i * 64 + m];
        float xi = xfb[i * 64 + 32 + m];
        float ar = wr[((size_t)i * W_ + o) * MODES_ + m];
        float ai = wi[((size_t)i * W_ + o) * MODES_ + m];
        zr += xr * ar - xi * ai;
        zi += xr * ai + xi * ar;
    }
    size_t row = (size_t)b * W_ + o;
    Z[row * 64 + m]      = zr;
    Z[row * 64 + 32 + m] = zi;
}

// ---------------- K4: fused inverse-DFT + pointwise conv + bias + GELU (WMMA) ----------------
// One wave computes ALL 64 output channels for one (b, t-tile): B-tiles of G and h
// are loaded once and reused across the 4 o-tiles (4x less h traffic; bandwidth-bound).
// hn[b,o,t] = gelu( sum_k Z[b,o,k]*G[k,t] + sum_i pw_w[o,i]*h[b,i,t] + pw_b[o] )
__global__ void fno_layer_kernel(const float* __restrict__ Z,
                                 const float* __restrict__ G,
                                 const float* __restrict__ h,
                                 const float* __restrict__ pw_w_l,  // [W,W]
                                 const float* __restrict__ pw_b_l,  // [W]
                                 float* __restrict__ hn) {
    int wave = (blockIdx.x * blockDim.x + threadIdx.x) >> 5;  // B*(T/16) = 16384 waves
    int lane = threadIdx.x & 31;
    int half = lane >> 4;
    int lid  = lane & 15;
    int b  = wave >> 8;
    int tt = wave & 255;
    int tBase = tt * 16;

    v8f c0 = {}, c1 = {}, c2 = {}, c3 = {};

    // part 1: inverse DFT. A tiles from Z rows, shared B tile from G.
    const float* zb = Z + (size_t)b * W_ * 64;
    for (int k = 0; k < 64; k += 4) {
        int kk = k + 2 * half;
        v2f bb;
        bb.x = G[(size_t)kk * T_ + tBase + lid];
        bb.y = G[(size_t)(kk + 1) * T_ + tBase + lid];
        v2f a0, a1, a2, a3;
        a0.x = zb[(      lid) * 64 + kk]; a0.y = zb[(      lid) * 64 + kk + 1];
        a1.x = zb[(16 +  lid) * 64 + kk]; a1.y = zb[(16 +  lid) * 64 + kk + 1];
        a2.x = zb[(32 +  lid) * 64 + kk]; a2.y = zb[(32 +  lid) * 64 + kk + 1];
        a3.x = zb[(48 +  lid) * 64 + kk]; a3.y = zb[(48 +  lid) * 64 + kk + 1];
        c0 = wmma4(a0, bb, c0);
        c1 = wmma4(a1, bb, c1);
        c2 = wmma4(a2, bb, c2);
        c3 = wmma4(a3, bb, c3);
    }
    // part 2: pointwise conv. A tiles from pw_w, shared B tile from h.
    const float* hb = h + (size_t)b * W_ * T_;
    for (int k = 0; k < 64; k += 4) {
        int kk = k + 2 * half;
        v2f bb;
        bb.x = hb[(size_t)kk * T_ + tBase + lid];
        bb.y = hb[(size_t)(kk + 1) * T_ + tBase + lid];
        v2f a0, a1, a2, a3;
        a0.x = pw_w_l[(      lid) * 64 + kk]; a0.y = pw_w_l[(      lid) * 64 + kk + 1];
        a1.x = pw_w_l[(16 +  lid) * 64 + kk]; a1.y = pw_w_l[(16 +  lid) * 64 + kk + 1];
        a2.x = pw_w_l[(32 +  lid) * 64 + kk]; a2.y = pw_w_l[(32 +  lid) * 64 + kk + 1];
        a3.x = pw_w_l[(48 +  lid) * 64 + kk]; a3.y = pw_w_l[(48 +  lid) * 64 + kk + 1];
        c0 = wmma4(a0, bb, c0);
        c1 = wmma4(a1, bb, c1);
        c2 = wmma4(a2, bb, c2);
        c3 = wmma4(a3, bb, c3);
    }
    #pragma unroll
    for (int r = 0; r < 8; ++r) {
        int ro = r + 8 * half;
        float v0 = c0[r] + pw_b_l[     ro];
        float v1 = c1[r] + pw_b_l[16 + ro];
        float v2 = c2[r] + pw_b_l[32 + ro];
        float v3 = c3[r] + pw_b_l[48 + ro];
        hn[((size_t)b * W_ +      ro) * T_ + tBase + lid] = gelu_exact(v0);
        hn[((size_t)b * W_ + 16 + ro) * T_ + tBase + lid] = gelu_exact(v1);
        hn[((size_t)b * W_ + 32 + ro) * T_ + tBase + lid] = gelu_exact(v2);
        hn[((size_t)b * W_ + 48 + ro) * T_ + tBase + lid] = gelu_exact(v3);
    }
}

// ---------------- K5: proj1 (WMMA) : p1[b,o,t] = gelu(proj1_w @ h + b1) ----------------
// Same full-channel-per-wave structure as K4 (h tile loaded once for all 4 o-tiles).
__global__ void fno_proj1_kernel(const float* __restrict__ h,
                                 const float* __restrict__ p1w,  // [W,W]
                                 const float* __restrict__ p1b,  // [W]
                                 float* __restrict__ p1) {
    int wave = (blockIdx.x * blockDim.x + threadIdx.x) >> 5;  // B*(T/16) = 16384 waves
    int lane = threadIdx.x & 31;
    int half = lane >> 4;
    int lid  = lane & 15;
    int b  = wave >> 8;
    int tt = wave & 255;
    int tBase = tt * 16;

    v8f c0 = {}, c1 = {}, c2 = {}, c3 = {};
    const float* hb = h + (size_t)b * W_ * T_;
    for (int k = 0; k < 64; k += 4) {
        int kk = k + 2 * half;
        v2f bb;
        bb.x = hb[(size_t)kk * T_ + tBase + lid];
        bb.y = hb[(size_t)(kk + 1) * T_ + tBase + lid];
        v2f a0, a1, a2, a3;
        a0.x = p1w[(      lid) * 64 + kk]; a0.y = p1w[(      lid) * 64 + kk + 1];
        a1.x = p1w[(16 +  lid) * 64 + kk]; a1.y = p1w[(16 +  lid) * 64 + kk + 1];
        a2.x = p1w[(32 +  lid) * 64 + kk]; a2.y = p1w[(32 +  lid) * 64 + kk + 1];
        a3.x = p1w[(48 +  lid) * 64 + kk]; a3.y = p1w[(48 +  lid) * 64 + kk + 1];
        c0 = wmma4(a0, bb, c0);
        c1 = wmma4(a1, bb, c1);
        c2 = wmma4(a2, bb, c2);
        c3 = wmma4(a3, bb, c3);
    }
    #pragma unroll
    for (int r = 0; r < 8; ++r) {
        int ro = r + 8 * half;
        float v0 = c0[r] + p1b[     ro];
        float v1 = c1[r] + p1b[16 + ro];
        float v2 = c2[r] + p1b[32 + ro];
        float v3 = c3[r] + p1b[48 + ro];
        p1[((size_t)b * W_ +      ro) * T_ + tBase + lid] = gelu_exact(v0);
        p1[((size_t)b * W_ + 16 + ro) * T_ + tBase + lid] = gelu_exact(v1);
        p1[((size_t)b * W_ + 32 + ro) * T_ + tBase + lid] = gelu_exact(v2);
        p1[((size_t)b * W_ + 48 + ro) * T_ + tBase + lid] = gelu_exact(v3);
    }
}

// ---------------- K6: proj2 reduction : s[b,t] = sum_o p1[b,o,t]*p2w[o] + p2b ----------------
__global__ void fno_proj2_kernel(const float* __restrict__ p1,
                                 const float* __restrict__ p2w,  // [W]
                                 const float* __restrict__ p2b,  // [1]
                                 float* __restrict__ s) {
    int idx = blockIdx.x * blockDim.x + threadIdx.x;   // B*T threads
    int t = idx & (T_ - 1);
    int b = idx >> 12;
    const float* pb = p1 + (size_t)b * W_ * T_ + t;
    float acc = p2b[0];
    for (int o = 0; o < W_; ++o)
        acc += pb[(size_t)o * T_] * p2w[o];
    s[(size_t)b * T_ + t] = acc;
}

// ---------------- K7: token matmul (WMMA) : out[b,u] = s[b,:] @ tok_w[u,:] + tok_b[u] ----------------
__global__ void fno_tok_kernel(const float* __restrict__ s,
                               const float* __restrict__ tw,   // [OUT_T, T]
                               const float* __restrict__ tb,   // [OUT_T]
                               float* __restrict__ out) {
    int wave = (blockIdx.x * blockDim.x + threadIdx.x) >> 5;  // 1024 waves
    int lane = threadIdx.x & 31;
    int half = lane >> 4;
    int lid  = lane & 15;
    int rowBase = (wave >> 8) * 16;   // batch rows (M=64)
    int colBase = (wave & 255) * 16;  // output tokens (N=4096)

    v8f acc = {};
    const float* arow = s + (size_t)(rowBase + lid) * T_;
    const float* brow = tw + (size_t)(colBase + lid) * T_;  // lane's own u-row (contiguous in k)
    for (int k = 0; k < T_; k += 4) {
        int kk = k + 2 * half;
        v2f a;  a.x = arow[kk];  a.y = arow[kk + 1];
        v2f bb; bb.x = brow[kk]; bb.y = brow[kk + 1];
        acc = wmma4(a, bb, acc);
    }
    #pragma unroll
    for (int r = 0; r < 8; ++r) {
        int row = rowBase + r + 8 * half;
        int col = colBase + lid;
        out[(size_t)row * OUT_T_ + col] = acc[r] + tb[col];
    }
}

// ---------------- launcher ----------------
extern "C" void kernel_launch(void* const* d_in, const int* in_sizes, int n_in,
                              void* d_out, int out_size, void* d_ws, size_t ws_size,
                              hipStream_t stream) {
    const float* x       = (const float*)d_in[0];
    const float* lift_w  = (const float*)d_in[1];
    const float* lift_b  = (const float*)d_in[2];
    const float* spec_wr = (const float*)d_in[3];
    const float* spec_wi = (const float*)d_in[4];
    const float* pw_w    = (const float*)d_in[5];
    const float* pw_b    = (const float*)d_in[6];
    const float* proj1_w = (const float*)d_in[7];
    const float* proj1_b = (const float*)d_in[8];
    const float* proj2_w = (const float*)d_in[9];
    const float* proj2_b = (const float*)d_in[10];
    const float* tok_w   = (const float*)d_in[11];
    const float* tok_b   = (const float*)d_in[12];
    float* out = (float*)d_out;

    // workspace carve (floats)
    float* F  = (float*)d_ws;                  // T*64
    float* G  = F  + (size_t)T_ * 64;          // 64*T
    float* XF = G  + (size_t)64 * T_;          // R*64
    float* Z  = XF + (size_t)R_ * 64;          // R*64
    float* S  = Z  + (size_t)R_ * 64;          // B*T
    float* hA = S  + (size_t)B_ * T_;          // B*W*T
    float* hB = hA + (size_t)B_ * W_ * T_;     // B*W*T

    const int TPB = 256;

    // K0: basis (recomputed each call; deterministic, cheap)
    fno_basis_kernel<<<(T_ * 64) / TPB, TPB, 0, stream>>>(F, G);

    // K1: lift
    fno_lift_kernel<<<(B_ * W_ * T_) / TPB, TPB, 0, stream>>>(x, lift_w, lift_b, hA);

    float* cur = hA;
    float* nxt = hB;
    for (int l = 0; l < NL_; ++l) {
        const float* wr_l  = spec_wr + (size_t)l * W_ * W_ * MODES_;
        const float* wi_l  = spec_wi + (size_t)l * W_ * W_ * MODES_;
        const float* pww_l = pw_w + (size_t)l * W_ * W_;
        const float* pwb_l = pw_b + (size_t)l * W_;

        // K2: forward DFT  (256 waves -> 32 blocks of 8 waves)
        fno_dft_kernel<<<(R_ / 16) / 8, TPB, 0, stream>>>(cur, F, XF);
        // K3: spectral mixing
        fno_specmul_kernel<<<(B_ * W_ * MODES_) / TPB, TPB, 0, stream>>>(XF, wr_l, wi_l, Z);
        // K4: fused inverse-DFT + pointwise + GELU  (16384 waves -> 2048 blocks)
        fno_layer_kernel<<<(B_ * (T_ / 16)) / 8, TPB, 0, stream>>>(Z, G, cur, pww_l, pwb_l, nxt);

        float* tmp = cur; cur = nxt; nxt = tmp;
    }

    // K5: proj1 + GELU  (16384 waves -> 2048 blocks)
    fno_proj1_kernel<<<(B_ * (T_ / 16)) / 8, TPB, 0, stream>>>(cur, proj1_w, proj1_b, nxt);
    // K6: proj2 reduce
    fno_proj2_kernel<<<(B_ * T_) / TPB, TPB, 0, stream>>>(nxt, proj2_w, proj2_b, S);
    // K7: token matmul  (1024 waves -> 128 blocks)
    fno_tok_kernel<<<((B_ / 16) * (OUT_T_ / 16)) / 8, TPB, 0, stream>>>(S, tok_w, tok_b, out);
}